// DenoiserWithMemoryAdapter_5317169512826
// MI455X (gfx1250) — compile-verified
//
#include <hip/hip_runtime.h>
#include <hip/hip_bf16.h>

typedef __attribute__((ext_vector_type(2))) float v2f;
typedef __attribute__((ext_vector_type(8))) float v8f;

// ---------------------------------------------------------------------------
// WMMA f32 16x16x4 wrapper (signature codegen-confirmed)
// ---------------------------------------------------------------------------
__device__ __forceinline__ v8f wmma_f32_16x16x4(v2f a, v2f b, v8f c) {
  return __builtin_amdgcn_wmma_f32_16x16x4_f32(false, a, false, b, (short)0, c,
                                               false, false);
}

// Order-preserving float -> uint32 key (smaller float -> smaller uint)
__device__ __forceinline__ unsigned int fkey(float f) {
  unsigned int u = __float_as_uint(f);
  return (u & 0x80000000u) ? ~u : (u | 0x80000000u);
}

// ---------------------------------------------------------------------------
// Shared conv2 (16ch -> 16ch, 3x3, relu) as implicit GEMM on WMMA.
// h1s: [36*36 pixels][stride 17] pixel-major input activations
// w2t: [co][tap*16+ci] (stride 145) transposed weights, K = tap*16+ci (144)
// h2s: [34*34 pixels][stride 17] pixel-major output
// Workgroup = 256 threads = 8 waves; each wave takes every 8th 16-pixel tile.
// ---------------------------------------------------------------------------
__device__ __forceinline__ void conv2_wmma(const float* __restrict__ h1s,
                                           const float* __restrict__ w2t,
                                           const float* __restrict__ bias,
                                           float* __restrict__ h2s) {
  const int t = threadIdx.x;
  const int wave = t >> 5;
  const int lane = t & 31;
  const int halfId = lane >> 4;
  const int r = lane & 15;

  for (int n = wave; n < 73; n += 8) {  // ceil(1156/16) tiles
    int p = n * 16 + r;
    bool valid = p < 1156;
    int pc = valid ? p : 1155;
    int y = pc / 34, x = pc % 34;
    int bases[9];
#pragma unroll
    for (int dy = 0; dy < 3; ++dy)
#pragma unroll
      for (int dx = 0; dx < 3; ++dx)
        bases[dy * 3 + dx] = ((y + dy) * 36 + (x + dx)) * 17;

    v8f acc;
#pragma unroll
    for (int j = 0; j < 8; ++j) acc[j] = bias[halfId * 8 + j];

    const float* arow = w2t + r * 145 + 2 * halfId;
#pragma unroll
    for (int kk = 0; kk < 144; kk += 4) {
      const int tap = kk >> 4;
      const int ci0 = (kk & 15) + 2 * halfId;
      const float* bp = &h1s[bases[tap] + ci0];
      v2f b; b.x = bp[0]; b.y = bp[1];
      const float* ap = arow + kk;
      v2f a; a.x = ap[0]; a.y = ap[1];
      acc = wmma_f32_16x16x4(a, b, acc);
    }
#pragma unroll
    for (int j = 0; j < 8; ++j) {
      float v = acc[j];
      if (valid) h2s[p * 17 + halfId * 8 + j] = v > 0.f ? v : 0.f;
    }
  }
}

// ---------------------------------------------------------------------------
// Kernel 0: init argmin slots
// ---------------------------------------------------------------------------
__global__ void init_slots_k(unsigned long long* __restrict__ slots) {
  int i = threadIdx.x;
  if (i < 512) slots[i] = 0xFFFFFFFFFFFFFFFFull;
}

// ---------------------------------------------------------------------------
// Kernel 1: fused base denoiser. conv1 (1->16) VALU, conv2 (16->16) WMMA,
// conv3 (16->1) VALU + residual. One 32x32 output tile per workgroup.
// ---------------------------------------------------------------------------
__global__ void base_fused_k(const float* __restrict__ noisy,
                             const float* __restrict__ w1, const float* __restrict__ b1,
                             const float* __restrict__ w2, const float* __restrict__ b2,
                             const float* __restrict__ w3, const float* __restrict__ b3,
                             float* __restrict__ base_out) {
  extern __shared__ float sm[];
  float* in_s = sm;                   // 38*38            = 1444
  float* h1s  = in_s + 1444;          // 1296*17          = 22032
  float* h2s  = h1s + 22032;          // 1156*17          = 19652
  float* w1s  = h2s + 19652;          // 144
  float* b1s  = w1s + 144;            // 16
  float* w2t  = b1s + 16;             // 16*145           = 2320
  float* b2s  = w2t + 2320;           // 16
  float* w3s  = b2s + 16;             // 144
  float* b3s  = w3s + 144;            // 1

  const int t = threadIdx.x;
  const int img = blockIdx.y;
  const int ty = (blockIdx.x >> 1) * 32;
  const int tx = (blockIdx.x & 1) * 32;
  const float* inp = noisy + (size_t)img * 4096;

  for (int i = t; i < 144; i += 256) { w1s[i] = w1[i]; w3s[i] = w3[i]; }
  for (int i = t; i < 16; i += 256) { b1s[i] = b1[i]; b2s[i] = b2[i]; }
  for (int i = t; i < 2304; i += 256) {  // transpose w2 -> [co][tap*16+ci]
    int co = i / 144, rem = i % 144, ci = rem / 9, tap = rem % 9;
    w2t[co * 145 + tap * 16 + ci] = w2[i];
  }
  if (t == 0) b3s[0] = b3[0];

  for (int i = t; i < 38 * 38; i += 256) {
    int y = i / 38, x = i % 38;
    int gy = ty + y - 3, gx = tx + x - 3;
    float v = 0.f;
    if (gy >= 0 && gy < 64 && gx >= 0 && gx < 64) v = inp[gy * 64 + gx];
    in_s[i] = v;
  }
  __syncthreads();

  // conv1: 1 -> 16, pixel-major output
  for (int i = t; i < 16 * 1296; i += 256) {
    int p = i >> 4, co = i & 15;
    int y = p / 36, x = p % 36;
    float v = b1s[co];
#pragma unroll
    for (int dy = 0; dy < 3; ++dy)
#pragma unroll
      for (int dx = 0; dx < 3; ++dx)
        v += w1s[co * 9 + dy * 3 + dx] * in_s[(y + dy) * 38 + (x + dx)];
    h1s[p * 17 + co] = v > 0.f ? v : 0.f;
  }
  __syncthreads();

  conv2_wmma(h1s, w2t, b2s, h2s);
  __syncthreads();

  // conv3 + residual
  for (int i = t; i < 1024; i += 256) {
    int y = i >> 5, x = i & 31;
    float v = b3s[0];
#pragma unroll
    for (int ci = 0; ci < 16; ++ci)
#pragma unroll
      for (int dy = 0; dy < 3; ++dy)
#pragma unroll
        for (int dx = 0; dx < 3; ++dx)
          v += w3s[ci * 9 + dy * 3 + dx] *
               h2s[((y + dy) * 34 + (x + dx)) * 17 + ci];
    int gy = ty + y, gx = tx + x;
    base_out[(size_t)img * 4096 + gy * 64 + gx] = inp[gy * 64 + gx] - v;
  }
}

// ---------------------------------------------------------------------------
// Kernel 2: distance GEMM + argmin.  key[b,n] = ||mf_n||^2 - 2*nf_b.mf_n
// 512 threads = 16 waves. Wave w: M rows [w*32, w*32+32) x N 64.
// Bank tile staged via global_load_async_to_lds_b128 (ASYNCcnt path).
// ---------------------------------------------------------------------------
#define KCHUNK 512
#define NT 64
#define LSTR 516  // padded row stride (floats) -> conflict-free ds reads

__global__ void dist_argmin_k(const float* __restrict__ nf,
                              const float* __restrict__ mf,
                              unsigned long long* __restrict__ slots) {
  extern __shared__ float sm[];
  float* tile = sm;                  // NT * LSTR
  float* b2s = sm + NT * LSTR;       // NT row-norm accumulators

  const int t = threadIdx.x;
  const int wave = t >> 5;           // 0..15
  const int lane = t & 31;
  const int halfId = lane >> 4;
  const int r = lane & 15;
  const int n0 = blockIdx.x * NT;

  if (t < NT) b2s[t] = 0.f;

  v8f acc[2][4];
#pragma unroll
  for (int i = 0; i < 2; ++i)
#pragma unroll
    for (int j = 0; j < 4; ++j) {
      v8f z = {};
      acc[i][j] = z;
    }

  for (int kc = 0; kc < 4096; kc += KCHUNK) {
    __syncthreads();  // previous chunk fully consumed (and b2s init visible)

    // ---- async-stage 64 rows x 512 cols into LDS (16 B per lane per op) ----
#pragma unroll
    for (int i = 0; i < 16; ++i) {
      int f = t + i * 512;
      int row = f >> 7;   // 128 float4 per row
      int c4 = f & 127;
      const float* gp = mf + (size_t)(n0 + row) * 4096 + kc + c4 * 4;
      unsigned int lds_addr =
          (unsigned int)(uintptr_t)(&tile[row * LSTR + c4 * 4]);
      unsigned long long ga = (unsigned long long)(uintptr_t)gp;
      asm volatile("global_load_async_to_lds_b128 %0, %1, off"
                   :: "v"(lds_addr), "v"(ga)
                   : "memory");
      if (kc + KCHUNK < 4096) __builtin_prefetch(gp + 512, 0, 1);
    }
    asm volatile("s_wait_asynccnt 0x0" ::: "memory");
    __syncthreads();

    // ---- accumulate ||row||^2 while the chunk is resident ----
    {
      int row = t >> 3, part = t & 7;  // 8 threads per row, 64 floats each
      const float* p = &tile[row * LSTR + part * 64];
      float s = 0.f;
#pragma unroll 8
      for (int c = 0; c < 64; ++c) s += p[c] * p[c];
      atomicAdd(&b2s[row], s);
    }

    // ---- WMMA over the chunk ----
    const float* arow = nf + (size_t)(wave * 32 + r) * 4096 + kc + 2 * halfId;
    for (int kk = 0; kk < KCHUNK; kk += 4) {
      v2f bf[4];
#pragma unroll
      for (int nt = 0; nt < 4; ++nt) {
        const float* sp = &tile[(nt * 16 + r) * LSTR + kk + 2 * halfId];
        v2f b; b.x = sp[0]; b.y = sp[1];
        bf[nt] = b;
      }
      v2f af[2];
#pragma unroll
      for (int mt = 0; mt < 2; ++mt) {
        const float* ap = arow + (size_t)(mt * 16) * 4096 + kk;
        v2f a; a.x = ap[0]; a.y = ap[1];
        af[mt] = a;
      }
#pragma unroll
      for (int mt = 0; mt < 2; ++mt)
#pragma unroll
        for (int nt = 0; nt < 4; ++nt)
          acc[mt][nt] = wmma_f32_16x16x4(af[mt], bf[nt], acc[mt][nt]);
    }
  }
  __syncthreads();  // b2s complete

  // ---- per-batch argmin over this workgroup's 64 candidate rows ----
#pragma unroll
  for (int mt = 0; mt < 2; ++mt) {
#pragma unroll
    for (int j = 0; j < 8; ++j) {
      float best = 3.0e38f;
      int bestn = n0;
#pragma unroll
      for (int nt = 0; nt < 4; ++nt) {
        float key = b2s[nt * 16 + r] - 2.0f * acc[mt][nt][j];
        int n = n0 + nt * 16 + r;
        if (key < best) { best = key; bestn = n; }
      }
#pragma unroll
      for (int m = 1; m <= 8; m <<= 1) {
        float ob = __shfl_xor(best, m, 32);
        int on = __shfl_xor(bestn, m, 32);
        if (ob < best) { best = ob; bestn = on; }
      }
      if (r == 0) {
        int b = wave * 32 + mt * 16 + halfId * 8 + j;
        unsigned long long key =
            ((unsigned long long)fkey(best) << 32) | (unsigned int)bestn;
        atomicMin(&slots[b], key);
      }
    }
  }
}

// ---------------------------------------------------------------------------
// Kernel 3: fused adapter. Gathers mem_clean[idx]; conv1 (3->16) VALU,
// conv2 (16->16) WMMA, conv3 (16->1) VALU; out = base_out + delta.
// ---------------------------------------------------------------------------
__global__ void adapter_fused_k(const float* __restrict__ noisy,
                                const float* __restrict__ base_out,
                                const float* __restrict__ mem_clean,
                                const unsigned long long* __restrict__ slots,
                                const float* __restrict__ w1, const float* __restrict__ b1,
                                const float* __restrict__ w2, const float* __restrict__ b2,
                                const float* __restrict__ w3, const float* __restrict__ b3,
                                float* __restrict__ out) {
  extern __shared__ float sm[];
  float* in_s = sm;                   // 3*1444           = 4332
  float* h1s  = in_s + 4332;          // 1296*17          = 22032
  float* h2s  = h1s + 22032;          // 1156*17          = 19652
  float* w1s  = h2s + 19652;          // 432
  float* b1s  = w1s + 432;            // 16
  float* w2t  = b1s + 16;             // 16*145           = 2320
  float* b2s  = w2t + 2320;           // 16
  float* w3s  = b2s + 16;             // 144
  float* b3s  = w3s + 144;            // 1

  const int t = threadIdx.x;
  const int img = blockIdx.y;
  const int ty = (blockIdx.x >> 1) * 32;
  const int tx = (blockIdx.x & 1) * 32;

  const float* inp = noisy + (size_t)img * 4096;
  const float* bo = base_out + (size_t)img * 4096;
  unsigned int idx = (unsigned int)(slots[img] & 0xFFFFFFFFull);
  const float* cl = mem_clean + (size_t)idx * 4096;

  for (int i = t; i < 432; i += 256) w1s[i] = w1[i];
  for (int i = t; i < 16; i += 256) { b1s[i] = b1[i]; b2s[i] = b2[i]; }
  for (int i = t; i < 2304; i += 256) {  // transpose w2 -> [co][tap*16+ci]
    int co = i / 144, rem = i % 144, ci = rem / 9, tap = rem % 9;
    w2t[co * 145 + tap * 16 + ci] = w2[i];
  }
  for (int i = t; i < 144; i += 256) w3s[i] = w3[i];
  if (t == 0) b3s[0] = b3[0];

  for (int i = t; i < 3 * 1444; i += 256) {
    int ch = i / 1444, p = i % 1444, y = p / 38, x = p % 38;
    int gy = ty + y - 3, gx = tx + x - 3;
    float v = 0.f;
    if (gy >= 0 && gy < 64 && gx >= 0 && gx < 64) {
      const float* src = (ch == 0) ? inp : (ch == 1 ? bo : cl);
      v = src[gy * 64 + gx];
    }
    in_s[i] = v;
  }
  __syncthreads();

  // conv1: 3 -> 16, pixel-major output
  for (int i = t; i < 16 * 1296; i += 256) {
    int p = i >> 4, co = i & 15;
    int y = p / 36, x = p % 36;
    float v = b1s[co];
#pragma unroll
    for (int ch = 0; ch < 3; ++ch)
#pragma unroll
      for (int dy = 0; dy < 3; ++dy)
#pragma unroll
        for (int dx = 0; dx < 3; ++dx)
          v += w1s[co * 27 + ch * 9 + dy * 3 + dx] *
               in_s[ch * 1444 + (y + dy) * 38 + (x + dx)];
    h1s[p * 17 + co] = v > 0.f ? v : 0.f;
  }
  __syncthreads();

  conv2_wmma(h1s, w2t, b2s, h2s);
  __syncthreads();

  // conv3 + add base_out
  for (int i = t; i < 1024; i += 256) {
    int y = i >> 5, x = i & 31;
    float v = b3s[0];
#pragma unroll
    for (int ci = 0; ci < 16; ++ci)
#pragma unroll
      for (int dy = 0; dy < 3; ++dy)
#pragma unroll
        for (int dx = 0; dx < 3; ++dx)
          v += w3s[ci * 9 + dy * 3 + dx] *
               h2s[((y + dy) * 34 + (x + dx)) * 17 + ci];
    int gy = ty + y, gx = tx + x;
    out[(size_t)img * 4096 + gy * 64 + gx] = bo[gy * 64 + gx] + v;
  }
}

// ---------------------------------------------------------------------------
// launch
// ---------------------------------------------------------------------------
extern "C" void kernel_launch(void* const* d_in, const int* in_sizes, int n_in,
                              void* d_out, int out_size, void* d_ws, size_t ws_size,
                              hipStream_t stream) {
  const float* noisy     = (const float*)d_in[0];
  const float* mem_noise = (const float*)d_in[1];
  const float* mem_clean = (const float*)d_in[2];
  const float* bw1 = (const float*)d_in[3];
  const float* bb1 = (const float*)d_in[4];
  const float* bw2 = (const float*)d_in[5];
  const float* bb2 = (const float*)d_in[6];
  const float* bw3 = (const float*)d_in[7];
  const float* bb3 = (const float*)d_in[8];
  const float* aw1 = (const float*)d_in[9];
  const float* ab1 = (const float*)d_in[10];
  const float* aw2 = (const float*)d_in[11];
  const float* ab2 = (const float*)d_in[12];
  const float* aw3 = (const float*)d_in[13];
  const float* ab3 = (const float*)d_in[14];
  float* out = (float*)d_out;

  unsigned long long* slots = (unsigned long long*)d_ws;
  float* base_out = (float*)((char*)d_ws + 4096);

  init_slots_k<<<1, 512, 0, stream>>>(slots);

  size_t lds_base = (size_t)(1444 + 22032 + 19652 +
                             144 + 16 + 2320 + 16 + 144 + 1) * sizeof(float);
  base_fused_k<<<dim3(4, 512), 256, lds_base, stream>>>(
      noisy, bw1, bb1, bw2, bb2, bw3, bb3, base_out);

  size_t lds_dist = (size_t)(NT * LSTR + NT) * sizeof(float);
  dist_argmin_k<<<512, 512, lds_dist, stream>>>(noisy, mem_noise, slots);

  size_t lds_adpt = (size_t)(4332 + 22032 + 19652 +
                             432 + 16 + 2320 + 16 + 144 + 1) * sizeof(float);
  adapter_fused_k<<<dim3(4, 512), 256, lds_adpt, stream>>>(
      noisy, base_out, mem_clean, slots, aw1, ab1, aw2, ab2, aw3, ab3, out);
}